// EnergyGatedDeltaModel_3204045603467
// MI455X (gfx1250) — compile-verified
//
#include <hip/hip_runtime.h>

#define Ht 64
#define Lt 2048
#define Bt 256
#define VOC 64

#define ENC_GRID 1024     // persistent encode blocks
#define ENC_GROUPS 8192   // 32768 tiles / 4 tiles-per-group

typedef float v2f __attribute__((ext_vector_type(2)));
typedef float v8f __attribute__((ext_vector_type(8)));

// ---------------------------------------------------------------------------
// WMMA f32 16x16x4 helpers (CDNA5, wave32). D = A(16x4) * B(4x16) + C(16x16)
// A layout (ISA 7.12.2): lane l -> row = l&15, k = 2*(l>>4) + {0,1}
// B layout (C-analog striping): VGPR0: lanes0-15 K=kb+0, lanes16-31 K=kb+2;
//                               VGPR1: lanes0-15 K=kb+1, lanes16-31 K=kb+3
// C layout: VGPR r -> row = r + 8*(l>>4), col = l&15
// ---------------------------------------------------------------------------
__device__ __forceinline__ v8f wmma4(v2f a, v2f b, v8f c) {
  return __builtin_amdgcn_wmma_f32_16x16x4_f32(false, a, false, b, (short)0, c,
                                               false, false);
}

__device__ __forceinline__ v2f ldA(const float* X, int ld, int kbase, int lane) {
  int row = lane & 15;
  int kh  = (lane >> 4) << 1;
  v2f a;
  a.x = X[row * ld + kbase + kh];
  a.y = X[row * ld + kbase + kh + 1];
  return a;
}

__device__ __forceinline__ v2f ldB(const float* W, int ld, int kbase, int nbase,
                                   int lane) {
  int col = lane & 15;
  int kh  = (lane >> 4) << 1;
  v2f b;
  b.x = W[(kbase + kh    ) * ld + nbase + col];
  b.y = W[(kbase + kh + 1) * ld + nbase + col];
  return b;
}

// ---------------------------------------------------------------------------
// Phase 1: encode. hidden = LN(h + relu(h@W1+b1)@W2+b2), h = embed[seq]
// Persistent blocks: 128 threads = 4 waves, each wave one 16-token tile per
// group iteration; W1/W2/biases staged in LDS once per block.
// ---------------------------------------------------------------------------
__global__ __launch_bounds__(128) void encode_kernel(
    const int* __restrict__ seq, const float* __restrict__ embed,
    const float* __restrict__ W1, const float* __restrict__ b1,
    const float* __restrict__ W2, const float* __restrict__ b2,
    const float* __restrict__ gamma, const float* __restrict__ beta,
    float* __restrict__ hidden) {
  __shared__ float W1s[64 * 128];      // 32 KB
  __shared__ float W2s[128 * 64];      // 32 KB
  __shared__ float cs[320];            // b1[128] | b2[64] | gamma[64] | beta[64]
  __shared__ float Xs[4][16 * 65];     // per-wave h tile (padded stride 65)
  __shared__ float Ys[4][16 * 17];     // per-wave relu staging (stride 17)

  const int tid  = threadIdx.x;
  const int lane = tid & 31;
  const int w    = tid >> 5;

  for (int i = tid; i < 64 * 128; i += 128) W1s[i] = W1[i];
  for (int i = tid; i < 128 * 64; i += 128) W2s[i] = W2[i];
  if (tid < 128) cs[tid] = b1[tid];
  if (tid < 64) {
    cs[128 + tid] = b2[tid];
    cs[192 + tid] = gamma[tid];
    cs[256 + tid] = beta[tid];
  }

  const int colc = lane & 15;
  const int kh2  = lane >> 4;
  float* X = &Xs[w][0];
  float* Y = &Ys[w][0];

  for (int g = blockIdx.x; g < ENC_GROUPS; g += ENC_GRID) {
    const int tb = (g * 4 + w) * 16;   // first flat token of this wave's tile

    // Gather h = embed[seq] into LDS (16 x 64)
    for (int i = lane; i < 1024; i += 32) {
      int r = i >> 6, c = i & 63;
      int s = seq[tb + r];
      X[r * 65 + c] = embed[s * 64 + c];
    }
    __syncthreads();   // X visible (also covers W1s/W2s/cs on first pass)

    // acc2 = layer-2 accumulators (16 x 64), seeded with b2
    v8f acc2[4];
    #pragma unroll
    for (int on = 0; on < 4; on++) {
      float bv = cs[128 + on * 16 + colc];
      #pragma unroll
      for (int r = 0; r < 8; r++) acc2[on][r] = bv;
    }

    for (int n = 0; n < 8; n++) {   // layer-1 N chunks of 16 (N=128)
      float bv = cs[n * 16 + colc];
      v8f c;
      #pragma unroll
      for (int r = 0; r < 8; r++) c[r] = bv;
      #pragma unroll
      for (int ks = 0; ks < 16; ks++) {   // K = 64 in steps of 4
        v2f a  = ldA(X, 65, ks * 4, lane);
        v2f bf = ldB(W1s, 128, ks * 4, n * 16, lane);
        c = wmma4(a, bf, c);
      }
      __syncthreads();   // all waves done reading Ys from previous n
      #pragma unroll
      for (int r = 0; r < 8; r++) {       // relu + stage 16x16
        float v = c[r];
        Y[(r + 8 * kh2) * 17 + colc] = v > 0.f ? v : 0.f;
      }
      __syncthreads();
      // layer 2: accumulate this 16-wide K chunk into all 4 output tiles
      #pragma unroll
      for (int on = 0; on < 4; on++) {
        #pragma unroll
        for (int ka = 0; ka < 4; ka++) {
          v2f a2 = ldA(Y, 17, ka * 4, lane);
          v2f bf = ldB(W2s, 64, n * 16 + ka * 4, on * 16, lane);
          acc2[on] = wmma4(a2, bf, acc2[on]);
        }
      }
    }

    // x = h + ff (overwrite X), then layernorm
    __syncthreads();
    #pragma unroll
    for (int on = 0; on < 4; on++) {
      #pragma unroll
      for (int r = 0; r < 8; r++) {
        int row = r + 8 * kh2;
        X[row * 65 + on * 16 + colc] += acc2[on][r];
      }
    }
    __syncthreads();
    if (lane < 16) {   // lanes 0..15: per-row mean / rstd
      float mu = 0.f;
      for (int j = 0; j < 64; j++) mu += X[lane * 65 + j];
      mu *= (1.f / 64.f);
      float var = 0.f;
      for (int j = 0; j < 64; j++) {
        float d = X[lane * 65 + j] - mu;
        var += d * d;
      }
      var *= (1.f / 64.f);
      Y[lane]      = mu;
      Y[32 + lane] = 1.f / sqrtf(var + 1e-5f);
    }
    __syncthreads();
    for (int i = lane; i < 1024; i += 32) {
      int r = i >> 6, c = i & 63;
      float v = (X[r * 65 + c] - Y[r]) * Y[32 + r] * cs[192 + c] + cs[256 + c];
      hidden[(size_t)tb * 64 + i] = v;
    }
    __syncthreads();   // protect X/Y before next group iteration
  }
}

// ---------------------------------------------------------------------------
// Phase 2: energy-gated delta-memory scan. One 64-thread block per batch;
// thread i owns row i of M (64 f32 in VGPRs). 2047 sequential steps.
// Double-buffered LDS (2 barriers/step), k(t+1) prefetched, 4-way partial
// accumulators to shorten dependent-FMA chains. All reductions use an
// identical order on every thread so the gate decision is uniform.
// ---------------------------------------------------------------------------
__global__ __launch_bounds__(64) void scan_kernel(
    const float* __restrict__ hidden, float* __restrict__ ctx,
    float* __restrict__ wcnt) {
  __shared__ float  kb[2][64];
  __shared__ float2 red[2][64];

  const int tid = threadIdx.x;
  const int b   = blockIdx.x;
  const float* hb = hidden + (size_t)b * Lt * Ht;

  float M[64];
  #pragma unroll
  for (int j = 0; j < 64; j++) M[j] = 0.f;
  float row_sq = 0.f;        // sum_j M[tid][j]^2, kept incrementally
  int   local_writes = 0;

  float knext = hb[tid];     // k(0); at t=2046 this prefetches the query row

  for (int t = 0; t < Lt - 1; t++) {
    const int p = t & 1;
    const float kself = knext;
    kb[p][tid] = kself;
    knext = hb[(size_t)(t + 1) * 64 + tid];   // prefetch next step
    __syncthreads();                          // barrier 1: kb[p] visible

    float kr[64];
    #pragma unroll
    for (int j = 0; j < 64; j++) kr[j] = kb[p][j];

    float kk0 = 0.f, kk1 = 0.f, kk2 = 0.f, kk3 = 0.f;
    float vp0 = 0.f, vp1 = 0.f, vp2 = 0.f, vp3 = 0.f;
    #pragma unroll
    for (int j = 0; j < 64; j += 4) {
      kk0 = fmaf(kr[j    ], kr[j    ], kk0);
      kk1 = fmaf(kr[j + 1], kr[j + 1], kk1);
      kk2 = fmaf(kr[j + 2], kr[j + 2], kk2);
      kk3 = fmaf(kr[j + 3], kr[j + 3], kk3);
      vp0 = fmaf(M[j    ], kr[j    ], vp0);
      vp1 = fmaf(M[j + 1], kr[j + 1], vp1);
      vp2 = fmaf(M[j + 2], kr[j + 2], vp2);
      vp3 = fmaf(M[j + 3], kr[j + 3], vp3);
    }
    const float kk = (kk0 + kk1) + (kk2 + kk3);
    const float vp = (vp0 + vp1) + (vp2 + vp3);

    const float delta = kself - vp / (kk + 1e-6f);   // v = k

    float s0 = 0.f, s1 = 0.f, s2 = 0.f, s3 = 0.f;
    #pragma unroll
    for (int j = 0; j < 64; j += 4) {
      float m0 = fmaf(delta, kr[j    ], M[j    ]); s0 = fmaf(m0, m0, s0);
      float m1 = fmaf(delta, kr[j + 1], M[j + 1]); s1 = fmaf(m1, m1, s1);
      float m2 = fmaf(delta, kr[j + 2], M[j + 2]); s2 = fmaf(m2, m2, s2);
      float m3 = fmaf(delta, kr[j + 3], M[j + 3]); s3 = fmaf(m3, m3, s3);
    }
    const float snew = (s0 + s1) + (s2 + s3);

    red[p][tid] = make_float2(row_sq, snew);
    __syncthreads();                          // barrier 2: red[p] visible

    float eo0 = 0.f, eo1 = 0.f, en0 = 0.f, en1 = 0.f;
    #pragma unroll
    for (int j = 0; j < 64; j += 2) {
      float2 a = red[p][j];
      float2 q = red[p][j + 1];
      eo0 += a.x; en0 += a.y;
      eo1 += q.x; en1 += q.y;
    }
    const float eo = eo0 + eo1, en = en0 + en1;

    const bool gate = (-0.5f * en) < (-0.5f * eo);   // E_after < E_before
    if (gate) {
      #pragma unroll
      for (int j = 0; j < 64; j++) M[j] = fmaf(delta, kr[j], M[j]);
      row_sq = snew;
      local_writes++;
    }
    // no trailing barrier: next step writes the other parity buffers
  }

  // ctx = M @ query; the query row hidden[:, L-1, :] is already in knext
  kb[0][tid] = knext;
  __syncthreads();
  float c0 = 0.f, c1 = 0.f, c2 = 0.f, c3 = 0.f;
  #pragma unroll
  for (int j = 0; j < 64; j += 4) {
    c0 = fmaf(M[j    ], kb[0][j    ], c0);
    c1 = fmaf(M[j + 1], kb[0][j + 1], c1);
    c2 = fmaf(M[j + 2], kb[0][j + 2], c2);
    c3 = fmaf(M[j + 3], kb[0][j + 3], c3);
  }
  ctx[b * 64 + tid] = (c0 + c1) + (c2 + c3);

  if (tid == 0) atomicAdd(wcnt, (float)local_writes);
}

// ---------------------------------------------------------------------------
// Phase 3: logits = (ctx @ Wr + br) @ Wo + bo, plus write_rate scalar.
// 4 blocks x 128 threads; each wave handles 16 rows of ctx.
// ---------------------------------------------------------------------------
__global__ __launch_bounds__(128) void head_kernel(
    const float* __restrict__ ctx, const float* __restrict__ Wr,
    const float* __restrict__ br, const float* __restrict__ Wo,
    const float* __restrict__ bo, const float* __restrict__ wcnt,
    float* __restrict__ out) {
  __shared__ float Wrs[64 * 64];    // 16 KB
  __shared__ float Wos[64 * 64];    // 16 KB
  __shared__ float bs[128];         // br | bo
  __shared__ float Xs[4][16 * 65];
  __shared__ float Ys[4][16 * 17];

  const int tid  = threadIdx.x;
  const int lane = tid & 31;
  const int w    = tid >> 5;

  for (int i = tid; i < 64 * 64; i += 128) { Wrs[i] = Wr[i]; Wos[i] = Wo[i]; }
  if (tid < 64) { bs[tid] = br[tid]; bs[64 + tid] = bo[tid]; }

  const int rowbase = (blockIdx.x * 4 + w) * 16;
  float* X = &Xs[w][0];
  float* Y = &Ys[w][0];

  for (int i = lane; i < 1024; i += 32) {
    int r = i >> 6, c = i & 63;
    X[r * 65 + c] = ctx[(rowbase + r) * 64 + c];
  }
  __syncthreads();

  const int colc = lane & 15;
  const int kh2  = lane >> 4;

  v8f acc2[4];
  #pragma unroll
  for (int on = 0; on < 4; on++) {
    float bv = bs[64 + on * 16 + colc];
    #pragma unroll
    for (int r = 0; r < 8; r++) acc2[on][r] = bv;
  }

  for (int n = 0; n < 4; n++) {     // tmp N chunks of 16 (N=64)
    float bv = bs[n * 16 + colc];
    v8f c;
    #pragma unroll
    for (int r = 0; r < 8; r++) c[r] = bv;
    #pragma unroll
    for (int ks = 0; ks < 16; ks++) {
      v2f a  = ldA(X, 65, ks * 4, lane);
      v2f bf = ldB(Wrs, 64, ks * 4, n * 16, lane);
      c = wmma4(a, bf, c);
    }
    __syncthreads();
    #pragma unroll
    for (int r = 0; r < 8; r++) Y[(r + 8 * kh2) * 17 + colc] = c[r];
    __syncthreads();
    #pragma unroll
    for (int on = 0; on < 4; on++) {
      #pragma unroll
      for (int ka = 0; ka < 4; ka++) {
        v2f a2 = ldA(Y, 17, ka * 4, lane);
        v2f bf = ldB(Wos, 64, n * 16 + ka * 4, on * 16, lane);
        acc2[on] = wmma4(a2, bf, acc2[on]);
      }
    }
  }

  #pragma unroll
  for (int on = 0; on < 4; on++) {
    #pragma unroll
    for (int r = 0; r < 8; r++) {
      int row = rowbase + r + 8 * kh2;
      out[row * 64 + on * 16 + colc] = acc2[on][r];
    }
  }

  if (blockIdx.x == 0 && tid == 0)
    out[Bt * VOC] = wcnt[0] * (1.0f / (float)(Bt * (Lt - 1)));
}

__global__ void zero1_kernel(float* p) { p[0] = 0.f; }

// ---------------------------------------------------------------------------
extern "C" void kernel_launch(void* const* d_in, const int* in_sizes, int n_in,
                              void* d_out, int out_size, void* d_ws,
                              size_t ws_size, hipStream_t stream) {
  (void)in_sizes; (void)n_in; (void)out_size; (void)ws_size;
  const int*   seq   = (const int*)  d_in[0];
  const float* embed = (const float*)d_in[1];
  const float* W1    = (const float*)d_in[2];
  const float* b1    = (const float*)d_in[3];
  const float* W2    = (const float*)d_in[4];
  const float* b2    = (const float*)d_in[5];
  const float* gamma = (const float*)d_in[6];
  const float* beta  = (const float*)d_in[7];
  const float* Wr    = (const float*)d_in[8];
  const float* br    = (const float*)d_in[9];
  const float* Wo    = (const float*)d_in[10];
  const float* bo    = (const float*)d_in[11];
  float* out = (float*)d_out;

  float* ws     = (float*)d_ws;
  float* hidden = ws;                              // B*L*H = 33,554,432 f32
  float* ctx    = ws + (size_t)Bt * Lt * Ht;       // B*H
  float* wcnt   = ctx + (size_t)Bt * Ht;           // 1 f32

  zero1_kernel<<<dim3(1), dim3(1), 0, stream>>>(wcnt);
  encode_kernel<<<dim3(ENC_GRID), dim3(128), 0, stream>>>(
      seq, embed, W1, b1, W2, b2, gamma, beta, hidden);
  scan_kernel<<<dim3(256), dim3(64), 0, stream>>>(hidden, ctx, wcnt);
  head_kernel<<<dim3(4), dim3(128), 0, stream>>>(ctx, Wr, br, Wo, bo, wcnt, out);
}